// TAG_40054865003184
// MI455X (gfx1250) — compile-verified
//
#include <hip/hip_runtime.h>
#include <hip/hip_bf16.h>

#define N_NODES 100000
#define N_EDGES 1600000
#define N_GRAPHS 64
#define FEAT 64
#define KHOPS 3
#define C_OUTS 5

typedef float v2f __attribute__((ext_vector_type(2)));
typedef float v8f __attribute__((ext_vector_type(8)));

#if defined(__HIP_DEVICE_COMPILE__)
#if !__has_builtin(__builtin_amdgcn_wmma_f32_16x16x4_f32)
#error "gfx1250 wmma f32 builtin missing"
#endif
#endif

// ---------------- utility kernels ----------------

__global__ void zerof_kernel(float* __restrict__ p, int n) {
  int i = blockIdx.x * blockDim.x + threadIdx.x;
  if (i < n) p[i] = 0.0f;
}

__global__ void deg_kernel(const int* __restrict__ dst, float* __restrict__ deg) {
  int e = blockIdx.x * blockDim.x + threadIdx.x;
  if (e < N_EDGES) atomicAdd(&deg[dst[e]], 1.0f);
}

__global__ void dinv_kernel(const float* __restrict__ deg, float* __restrict__ dinv) {
  int i = blockIdx.x * blockDim.x + threadIdx.x;
  if (i < N_NODES) {
    float d = deg[i];
    dinv[i] = (d > 0.0f) ? rsqrtf(fmaxf(d, 1.0f)) : 0.0f;
  }
}

__global__ void norm_kernel(const int* __restrict__ src, const int* __restrict__ dst,
                            const float* __restrict__ dinv, float* __restrict__ nrm) {
  int e = blockIdx.x * blockDim.x + threadIdx.x;
  if (e < N_EDGES) nrm[e] = dinv[src[e]] * dinv[dst[e]];
}

// ---------------- sparse propagation: hout[dst] += h[src] * norm ----------------
// 16 threads per edge, float4 per thread. L2-resident gather + f32 atomics.

__global__ void propagate_kernel(const float* __restrict__ h,
                                 const float* __restrict__ nrm,
                                 const int* __restrict__ src,
                                 const int* __restrict__ dst,
                                 float* __restrict__ hout) {
  int tid = blockIdx.x * blockDim.x + threadIdx.x;   // < N_EDGES*16
  int e  = tid >> 4;
  int f4 = (tid & 15) << 2;
  if (e >= N_EDGES) return;
  if ((tid & 1023) == 0) {
    // keep edge streams flowing (global_prefetch_b8)
    __builtin_prefetch(src + e + 4096, 0, 1);
    __builtin_prefetch(dst + e + 4096, 0, 1);
    __builtin_prefetch(nrm + e + 4096, 0, 1);
  }
  float nv = nrm[e];
  int s = src[e];
  int d = dst[e];
  const float4 v = *(const float4*)(h + (size_t)s * FEAT + f4);
  float* o = hout + (size_t)d * FEAT + f4;
  atomicAdd(o + 0, v.x * nv);
  atomicAdd(o + 1, v.y * nv);
  atomicAdd(o + 2, v.z * nv);
  atomicAdd(o + 3, v.w * nv);
}

// ---------------- dense GEMM via WMMA f32 16x16x4 ----------------
// C[N,64] = (init ? bias : C) + A[N,64] @ W[64,64]
// 128-thread block = 4 waves; wave w does rows [16*bid,16*bid+16), cols [16w,16w+16).
// A operand (16x4 f32, 2 VGPRs): lane m=lane&15 holds A[m][k0+2*(lane>>4) + {0,1}].
// B operand symmetric with N in place of M. C/D: 8 VGPRs, elem (j+8*(lane>>4), lane&15).

__global__ void gemm64_wmma_kernel(const float* __restrict__ A,
                                   const float* __restrict__ W,
                                   const float* __restrict__ bias,
                                   float* __restrict__ C,
                                   int init) {
  const int wave = threadIdx.x >> 5;
  const int lane = threadIdx.x & 31;
  const int half = lane >> 4;      // 0: K/M low half, 1: high half
  const int lid  = lane & 15;
  const int row0 = blockIdx.x * 16;
  const int c0   = wave * 16;

  v8f acc;
  if (init) {
    float bv = (bias != nullptr) ? bias[c0 + lid] : 0.0f;
#pragma unroll
    for (int j = 0; j < 8; ++j) acc[j] = bv;
  } else {
#pragma unroll
    for (int j = 0; j < 8; ++j)
      acc[j] = C[(size_t)(row0 + j + 8 * half) * FEAT + c0 + lid];
  }

  const float* Arow = A + (size_t)(row0 + lid) * FEAT;

#pragma unroll
  for (int k0 = 0; k0 < FEAT; k0 += 4) {
    const int kk = k0 + half * 2;
    v2f a, b;
    a.x = Arow[kk];
    a.y = Arow[kk + 1];
    b.x = W[(size_t)kk * FEAT + c0 + lid];
    b.y = W[(size_t)(kk + 1) * FEAT + c0 + lid];
    acc = __builtin_amdgcn_wmma_f32_16x16x4_f32(
        /*neg_a=*/false, a, /*neg_b=*/false, b,
        /*c_mod=*/(short)0, acc, /*reuse_a=*/false, /*reuse_b=*/false);
  }

#pragma unroll
  for (int j = 0; j < 8; ++j)
    C[(size_t)(row0 + j + 8 * half) * FEAT + c0 + lid] = acc[j];
}

// ---------------- global mean pool (accumulate) ----------------

__global__ void pool_kernel(const float* __restrict__ h,
                            const int* __restrict__ batch,
                            float* __restrict__ pool) {  // [G*64 sums][G cnt]
  int tid = blockIdx.x * blockDim.x + threadIdx.x;   // < N_NODES*16
  int n  = tid >> 4;
  int f4 = (tid & 15) << 2;
  if (n >= N_NODES) return;
  int g = batch[n];
  const float4 v = *(const float4*)(h + (size_t)n * FEAT + f4);
  float* o = pool + (size_t)g * FEAT + f4;
  atomicAdd(o + 0, v.x);
  atomicAdd(o + 1, v.y);
  atomicAdd(o + 2, v.z);
  atomicAdd(o + 3, v.w);
  if ((tid & 15) == 0) atomicAdd(&pool[N_GRAPHS * FEAT + g], 1.0f);
}

// ---------------- final tiny linear: out[g,c] = mean(g) @ Wf + bf ----------------

__global__ void final_kernel(const float* __restrict__ pool,
                             const float* __restrict__ Wf,
                             const float* __restrict__ bf,
                             float* __restrict__ out) {
  int g = threadIdx.x;
  if (g >= N_GRAPHS) return;
  float inv = 1.0f / fmaxf(pool[N_GRAPHS * FEAT + g], 1.0f);
  float accs[C_OUTS];
#pragma unroll
  for (int c = 0; c < C_OUTS; ++c) accs[c] = bf[c];
  for (int f = 0; f < FEAT; ++f) {
    float hv = pool[g * FEAT + f] * inv;
#pragma unroll
    for (int c = 0; c < C_OUTS; ++c) accs[c] += hv * Wf[f * C_OUTS + c];
  }
#pragma unroll
  for (int c = 0; c < C_OUTS; ++c) out[g * C_OUTS + c] = accs[c];
}

// ---------------- host orchestration ----------------

extern "C" void kernel_launch(void* const* d_in, const int* in_sizes, int n_in,
                              void* d_out, int out_size, void* d_ws, size_t ws_size,
                              hipStream_t stream) {
  const float* x    = (const float*)d_in[0];
  const float* W_in = (const float*)d_in[1];
  const float* b_in = (const float*)d_in[2];
  const float* W_h  = (const float*)d_in[3];
  const float* b_h  = (const float*)d_in[4];
  const float* W_f  = (const float*)d_in[5];
  const float* b_f  = (const float*)d_in[6];
  const int*   ei   = (const int*)d_in[7];
  const int*   bat  = (const int*)d_in[8];
  const int* srcp = ei;             // edge_index row 0
  const int* dstp = ei + N_EDGES;   // edge_index row 1

  float* ws   = (float*)d_ws;
  float* deg  = ws;
  float* dinv = deg + N_NODES;
  float* nrm  = dinv + N_NODES;
  float* B[4];
  B[0] = nrm + N_EDGES;
  for (int i = 1; i < 4; ++i) B[i] = B[i - 1] + (size_t)N_NODES * FEAT;
  float* pool = B[3] + (size_t)N_NODES * FEAT;  // G*64 sums + G counts

  const int T = 256;
  const int NF = N_NODES * FEAT;

  // gcn_norm
  zerof_kernel<<<(N_NODES + T - 1) / T, T, 0, stream>>>(deg, N_NODES);
  deg_kernel<<<(N_EDGES + T - 1) / T, T, 0, stream>>>(dstp, deg);
  dinv_kernel<<<(N_NODES + T - 1) / T, T, 0, stream>>>(deg, dinv);
  norm_kernel<<<(N_EDGES + T - 1) / T, T, 0, stream>>>(srcp, dstp, dinv, nrm);

  auto conv = [&](const float* Xin, const float* Wm, const float* bias,
                  float* acc, float* h0, float* h1) {
    gemm64_wmma_kernel<<<N_NODES / 16, 128, 0, stream>>>(Xin, Wm, bias, acc, 1);
    const float* cur = Xin;
    float* nxt = h0;
    for (int k = 1; k <= KHOPS; ++k) {
      zerof_kernel<<<(NF + T - 1) / T, T, 0, stream>>>(nxt, NF);
      propagate_kernel<<<(N_EDGES * 16) / T, T, 0, stream>>>(cur, nrm, srcp, dstp, nxt);
      gemm64_wmma_kernel<<<N_NODES / 16, 128, 0, stream>>>(
          nxt, Wm + (size_t)k * FEAT * FEAT, nullptr, acc, 0);
      cur = nxt;
      nxt = (nxt == h0) ? h1 : h0;
    }
  };

  // input conv + 3 hidden convs (buffer rotation keeps acc/input/hops disjoint)
  conv(x,    W_in,                          b_in,        B[0], B[1], B[2]);
  conv(B[0], W_h + (size_t)0 * 4 * FEAT * FEAT, b_h + 0 * FEAT, B[1], B[2], B[3]);
  conv(B[1], W_h + (size_t)1 * 4 * FEAT * FEAT, b_h + 1 * FEAT, B[2], B[3], B[0]);
  conv(B[2], W_h + (size_t)2 * 4 * FEAT * FEAT, b_h + 2 * FEAT, B[3], B[0], B[1]);

  // global mean pool + final linear
  zerof_kernel<<<1, 256, 0, stream>>>(pool, N_GRAPHS * (FEAT + 1));
  pool_kernel<<<(N_NODES * 16 + T - 1) / T, T, 0, stream>>>(B[3], bat, pool);
  final_kernel<<<1, 64, 0, stream>>>(pool, W_f, b_f, (float*)d_out);
}